// GCNEncoder_58789512347871
// MI455X (gfx1250) — compile-verified
//
#include <hip/hip_runtime.h>

#define B_ 2048
#define N_ 100
#define D_ 128
#define L_ 3

typedef float v2f __attribute__((ext_vector_type(2)));
typedef float v8f __attribute__((ext_vector_type(8)));

// ---------------- Stage 1: per-batch mean of locs (one wave32 per batch) ----
__global__ __launch_bounds__(256) void mean_kernel(const float* __restrict__ locs,
                                                   float* __restrict__ meanloc) {
    int gid  = blockIdx.x * blockDim.x + threadIdx.x;
    int b    = gid >> 5;
    int lane = gid & 31;
    float sx = 0.f, sy = 0.f;
    for (int n = lane; n < N_; n += 32) {
        const float* p = locs + (b * N_ + n) * 2;
        sx += p[0];
        sy += p[1];
    }
#pragma unroll
    for (int off = 16; off >= 1; off >>= 1) {
        sx += __shfl_xor(sx, off, 32);
        sy += __shfl_xor(sy, off, 32);
    }
    if (lane == 0) {
        meanloc[b * 2 + 0] = sx * (1.0f / N_);
        meanloc[b * 2 + 1] = sy * (1.0f / N_);
    }
}

// ---------------- Stage 2: transpose Ws -> Wt[l][n][k] (L2-resident) -------
__global__ __launch_bounds__(256) void transpose_kernel(const float* __restrict__ Ws,
                                                        float* __restrict__ Wt) {
    int i = blockIdx.x * 256 + threadIdx.x;   // output linear index
    int l = i >> 14;
    int n = (i >> 7) & 127;
    int k = i & 127;
    Wt[i] = Ws[l * (D_ * D_) + k * D_ + n];
}

// ---------------- Stage 3: fp32 WMMA GEMM chain on [B,128] ------------------
// One wave owns a 16-row batch tile; activations live in LDS (8 KB per wave).
__global__ __launch_bounds__(128) void chain_kernel(const float* __restrict__ meanloc,
                                                    const float* __restrict__ W_init,
                                                    const float* __restrict__ b_init,
                                                    const float* __restrict__ Wt,  // [3][128][128], Wt[l][n][k]
                                                    const float* __restrict__ bs,  // [3][128]
                                                    float* __restrict__ g2) {      // [B][128]
    __shared__ float act[4][16 * D_];
    const int w    = threadIdx.x >> 5;
    const int lane = threadIdx.x & 31;
    const int lid  = lane & 15;   // M / N index within half-wave
    const int hs   = lane >> 4;   // half select (K pair / row+8)
    const int b0   = (blockIdx.x * 4 + w) * 16;

    // init embedding of the mean location: m = meanloc @ W_init + b_init
    for (int r = 0; r < 16; ++r) {
        float mx = meanloc[(b0 + r) * 2 + 0];
        float my = meanloc[(b0 + r) * 2 + 1];
        for (int d = lane; d < D_; d += 32) {
            act[w][r * D_ + d] = mx * W_init[d] + my * W_init[D_ + d] + b_init[d];
        }
    }
    __syncthreads();

    for (int layer = 0; layer < L_; ++layer) {
        const float* wt   = Wt + layer * D_ * D_;
        const float* bias = bs + layer * D_;

        v8f acc[8];
#pragma unroll
        for (int nt = 0; nt < 8; ++nt) acc[nt] = (v8f){0.f, 0.f, 0.f, 0.f, 0.f, 0.f, 0.f, 0.f};

        // D = A(16xK) * B(Kx16) with K stepped by 4 (V_WMMA_F32_16X16X4_F32).
        // A layout: lanes 0-15 hold M=lid, K = k0,k0+1; lanes 16-31 K = k0+2,k0+3.
        // B layout (via transposed Wt): lane holds column n = nt*16+lid, same K pair.
        for (int k0 = 0; k0 < D_; k0 += 4) {
            const int ks = k0 + 2 * hs;
            v2f av = *(const v2f*)(&act[w][lid * D_ + ks]);
#pragma unroll
            for (int nt = 0; nt < 8; ++nt) {
                v2f bv = *(const v2f*)(&wt[(nt * 16 + lid) * D_ + ks]);
                acc[nt] = __builtin_amdgcn_wmma_f32_16x16x4_f32(
                    false, av, false, bv, (short)0, acc[nt], false, false);
            }
        }

        // Epilogue: C/D layout — VGPR r holds row r (lanes 0-15) / row r+8 (lanes 16-31).
        if (layer < L_ - 1) {
#pragma unroll
            for (int nt = 0; nt < 8; ++nt) {
                int col = nt * 16 + lid;
                float bv = bias[col];
#pragma unroll
                for (int r = 0; r < 8; ++r) {
                    float v = acc[nt][r] + bv;
                    v = v > 0.f ? v : 0.f;   // relu
                    act[w][(r + hs * 8) * D_ + col] = v;
                }
            }
        } else {
#pragma unroll
            for (int nt = 0; nt < 8; ++nt) {
                int col = nt * 16 + lid;
                float bv = bias[col];
#pragma unroll
                for (int r = 0; r < 8; ++r) {
                    g2[(b0 + r + hs * 8) * D_ + col] = acc[nt][r] + bv;
                }
            }
        }
        __syncthreads();
    }
}

// ---------------- Stage 4: streaming writer (bandwidth-bound, ~210 MB) ------
// 32 consecutive lanes write one 512B row of h and of init_h (float4 stores).
__global__ __launch_bounds__(256) void bcast_kernel(const float* __restrict__ locs,
                                                    const float* __restrict__ W_init,
                                                    const float* __restrict__ b_init,
                                                    const float* __restrict__ g2,
                                                    float* __restrict__ out) {
    int idx = blockIdx.x * 256 + threadIdx.x;  // < B*N*32
    int row = idx >> 5;                        // b*N + n
    int dv  = idx & 31;
    int b   = row / N_;
    int d0  = dv * 4;

    float lx = locs[row * 2 + 0];
    float ly = locs[row * 2 + 1];
    float4 w0 = *(const float4*)(W_init + d0);
    float4 w1 = *(const float4*)(W_init + D_ + d0);
    float4 bi = *(const float4*)(b_init + d0);
    float4 g  = *(const float4*)(g2 + b * D_ + d0);

    float4 ih, h;
    ih.x = lx * w0.x + ly * w1.x + bi.x;
    ih.y = lx * w0.y + ly * w1.y + bi.y;
    ih.z = lx * w0.z + ly * w1.z + bi.z;
    ih.w = lx * w0.w + ly * w1.w + bi.w;
    h.x = ih.x + g.x;
    h.y = ih.y + g.y;
    h.z = ih.z + g.z;
    h.w = ih.w + g.w;

    int o = row * D_ + d0;
    *(float4*)(out + o) = h;
    *(float4*)(out + B_ * N_ * D_ + o) = ih;
}

extern "C" void kernel_launch(void* const* d_in, const int* in_sizes, int n_in,
                              void* d_out, int out_size, void* d_ws, size_t ws_size,
                              hipStream_t stream) {
    const float* locs   = (const float*)d_in[0];
    const float* W_init = (const float*)d_in[1];
    const float* b_init = (const float*)d_in[2];
    const float* Ws     = (const float*)d_in[3];
    const float* bs     = (const float*)d_in[4];
    float* out = (float*)d_out;

    char*  ws      = (char*)d_ws;
    float* meanloc = (float*)(ws);                   // 2048*2*4   = 16 KB
    float* Wt      = (float*)(ws + 16384);           // 3*128*128*4 = 192 KB
    float* g2      = (float*)(ws + 16384 + 196608);  // 2048*128*4  = 1 MB

    mean_kernel<<<B_ / 8, 256, 0, stream>>>(locs, meanloc);
    transpose_kernel<<<(L_ * D_ * D_) / 256, 256, 0, stream>>>(Ws, Wt);
    chain_kernel<<<B_ / 64, 128, 0, stream>>>(meanloc, W_init, b_init, Wt, bs, g2);
    bcast_kernel<<<(B_ * N_ * 32) / 256, 256, 0, stream>>>(locs, W_init, b_init, g2, out);
}